// DecodePredictions_51960514347098
// MI455X (gfx1250) — compile-verified
//
#include <hip/hip_runtime.h>
#include <hip/hip_bf16.h>
#include <stdint.h>

// ---------------------------------------------------------------------------
// RetinaNet DecodePredictions for MI455X (gfx1250, wave32).
// Memory-bound op (206MB predictions, 23.3TB/s HBM => ~9us/pass floor).
// No matmul => no WMMA. CDNA5 path: Tensor Data Mover streams contiguous
// 336B-per-anchor tiles into LDS (big DMA rows => request-rate efficient),
// double buffered with s_wait_tensorcnt. Exact top-100 per (b,class) via
// global coarse 8-bit radix + per-class candidate lists + in-LDS 2-level
// radix refinement; wave32 shuffle reductions for the tails.
// ---------------------------------------------------------------------------

#define NBATCH   8
#define NANCH    76725
#define NCLS     80
#define NFIELD   84          // 4 box + 80 class logits
#define KPER     100         // MAX_PER_CLASS
#define MAXDET   100
#define SLABS    32          // anchor slabs per image (grid = 8*32 = 256)
#define SLABR    2398        // ceil(76725/32)
#define CROWS    48          // anchor rows per TDM chunk (48*84*4B = 16128B)
#define CHUNKF   (CROWS * NFIELD)
#define NBINS    2048        // 11-bit radix histogram (fine stages)
#define BINCAP2  2048        // kernel-5 threshold-bin capacity
#define LISTCAP  2048        // per-(b,c) global candidate list capacity
#define FCAP     256         // fine-bin capacity
#define LOGIT_TH  (-2.9444389791664403f)   // logit(0.05)
#define REAL_MIN  (-1.0e30f)
#define PAD_VAL   (-3.0e38f)

#if defined(__has_builtin)
#  if __has_builtin(__builtin_amdgcn_tensor_load_to_lds) && __has_builtin(__builtin_amdgcn_s_wait_tensorcnt)
#    define USE_TDM 1
#  endif
#endif
#ifndef USE_TDM
#  define USE_TDM 0
#endif

typedef uint32_t u32x4 __attribute__((ext_vector_type(4)));
typedef int      i32x4 __attribute__((ext_vector_type(4)));
typedef int      i32x8 __attribute__((ext_vector_type(8)));

// Order-preserving float -> uint key (ascending key == ascending float).
__device__ __forceinline__ uint32_t fkey(float v) {
  uint32_t fb = __float_as_uint(v);
  return (fb & 0x80000000u) ? ~fb : (fb | 0x80000000u);
}

#if USE_TDM
// D# for a contiguous 1-row tile of `elems` 4B elements (elems <= 65535).
// Bit layout per CDNA5 ISA.
__device__ __forceinline__ void tdm_issue_tile(uint32_t lds_off, const float* gptr, int elems) {
  uint64_t ga = (uint64_t)(uintptr_t)gptr;
  u32x4 g0;
  g0[0] = 0x1u;                                              // count=1, user D#
  g0[1] = lds_off;                                           // lds_addr
  g0[2] = (uint32_t)(ga & 0xFFFFFFFFu);                      // global_addr lo
  g0[3] = (uint32_t)((ga >> 32) & 0x01FFFFFFu) | (2u << 30); // ga hi + type=2
  i32x8 g1;
  g1[0] = (int)(2u << 16);                                   // data_size=4B
  g1[1] = (int)(((uint32_t)elems & 0xFFFFu) << 16);          // tensor_dim0 lo16
  g1[2] = (int)(((uint32_t)elems >> 16) & 0xFFFFu) | (int)(1u << 16); // dim0 hi | tensor_dim1=1
  g1[3] = (int)(((uint32_t)elems & 0xFFFFu) << 16);          // tile_dim0 = elems
  g1[4] = 1;                                                 // tile_dim1 = 1
  g1[5] = (int)(uint32_t)elems;                              // dim0 stride (1 row; unused)
  g1[6] = 0;
  g1[7] = 0;
  i32x4 z4 = {0, 0, 0, 0};
#if __clang_major__ >= 23
  i32x8 z8 = {0, 0, 0, 0, 0, 0, 0, 0};
  __builtin_amdgcn_tensor_load_to_lds(g0, g1, z4, z4, z8, 0);
#else
  __builtin_amdgcn_tensor_load_to_lds(g0, g1, z4, z4, 0);
#endif
}
#endif

// Find T = max bin index such that count(bins >= T) >= need; -1 if total<need.
__device__ __forceinline__ int find_threshold_bin(unsigned int* hist, unsigned int* superS,
                                                  int need, int tid, int* slot) {
  if (tid < 64) {
    unsigned int s = 0;
    for (int k = 0; k < 32; ++k) s += hist[tid * 32 + k];
    superS[tid] = s;
  }
  __syncthreads();
  if (tid == 0) {
    int cum = 0, T = -1;
    for (int s = 63; s >= 0; --s) {
      int sv = (int)superS[s];
      if (sv > 0 && cum + sv >= need) {
        for (int i = s * 32 + 31; i >= s * 32; --i) {
          int h = (int)hist[i];
          if (h > 0 && cum + h >= need) { T = i; break; }
          cum += h;
        }
        break;
      }
      cum += sv;
    }
    *slot = T;
  }
  __syncthreads();
  return *slot;
}

// Wave32 shuffle argmax of (v, n, p): max v, tie -> min n.
__device__ __forceinline__ void wave_argmax3(float& v, int& n, int& p) {
  for (int m = 16; m > 0; m >>= 1) {
    float ov = __shfl_xor(v, m, 32);
    int   on = __shfl_xor(n, m, 32);
    int   op = __shfl_xor(p, m, 32);
    if (ov > v || (ov == v && on < n)) { v = ov; n = on; p = op; }
  }
}

// Append the exact top-`need` of threshold-bin list binS/binI[0..M) after G
// already-selected entries (fine 11-bit refinement + few pops). Returns total.
__device__ __forceinline__ int select_from_bin(float* binS, int* binI, int M, int G, int need,
                                               unsigned int* hist, unsigned int* superS,
                                               float* selS, int* selI,
                                               float* fS, int* fI,
                                               float* wS, int* wN, int* wP,
                                               int* ctrs, int tid) {
  __syncthreads();   // everyone has read G/M before counters are reused
  if (M <= need) {
    for (int i = tid; i < M; i += 256) { selS[G + i] = binS[i]; selI[G + i] = binI[i]; }
    __syncthreads();
    return G + M;
  }
  for (int i = tid; i < NBINS; i += 256) hist[i] = 0u;
  if (tid == 0) { ctrs[0] = 0; ctrs[1] = 0; }
  __syncthreads();
  for (int i = tid; i < M; i += 256)
    atomicAdd(&hist[(fkey(binS[i]) >> 10) & 0x7FFu], 1u);
  __syncthreads();
  const int T2 = find_threshold_bin(hist, superS, need, tid, &ctrs[2]);
  for (int i = tid; i < M; i += 256) {
    float v = binS[i]; int n = binI[i];
    int fbin = (int)((fkey(v) >> 10) & 0x7FFu);
    if (fbin > T2) {
      int p = atomicAdd(&ctrs[0], 1);
      selS[G + p] = v; selI[G + p] = n;      // p < need guaranteed
    } else if (fbin == T2) {
      int p = atomicAdd(&ctrs[1], 1);
      if (p < FCAP) { fS[p] = v; fI[p] = n; }
    }
  }
  __syncthreads();
  const int G2 = ctrs[0];
  int M2 = ctrs[1]; if (M2 > FCAP) M2 = FCAP;
  const int need2 = need - G2;
  int pc = 0;
  for (int r = 0; r < need2; ++r) {          // expected O(1) iterations
    float bv = PAD_VAL; int bn = 0x7FFFFFFF; int bp = -1;
    for (int i = tid; i < M2; i += 256) {
      float v = fS[i]; int n = fI[i];
      if (v > bv || (v == bv && n < bn)) { bv = v; bn = n; bp = i; }
    }
    wave_argmax3(bv, bn, bp);
    if ((tid & 31) == 0) { int w = tid >> 5; wS[w] = bv; wN[w] = bn; wP[w] = bp; }
    __syncthreads();
    if (tid == 0) {
      float v = wS[0]; int n = wN[0], p = wP[0];
      for (int w = 1; w < 8; ++w)
        if (wS[w] > v || (wS[w] == v && wN[w] < n)) { v = wS[w]; n = wN[w]; p = wP[w]; }
      if (p >= 0 && v > REAL_MIN) {
        selS[G + G2 + r] = v; selI[G + G2 + r] = n;
        fS[p] = PAD_VAL; ctrs[3] = 1;
      } else ctrs[3] = 0;
    }
    __syncthreads();
    if (!ctrs[3]) break;
    ++pc;
  }
  return G + G2 + pc;
}

// Bitonic sort of 128 (score, idx) entries: score desc, tie -> idx asc.
__device__ __forceinline__ void sort128_desc(float* selS, int* selI, int tid) {
  for (int kk = 2; kk <= 128; kk <<= 1) {
    for (int j = kk >> 1; j > 0; j >>= 1) {
      if (tid < 128) {
        int ixj = tid ^ j;
        if (ixj > tid) {
          float a = selS[tid], q = selS[ixj];
          int ai = selI[tid], qi = selI[ixj];
          bool aPre = (a > q) || (a == q && ai < qi);
          bool qPre = (q > a) || (q == a && qi < ai);
          bool asc = ((tid & kk) == 0);
          if (asc ? qPre : aPre) {
            selS[tid] = q; selS[ixj] = a;
            selI[tid] = qi; selI[ixj] = ai;
          }
        }
      }
      __syncthreads();
    }
  }
}

// K0: zero global histogram + list counters (ws is poisoned by the harness).
__global__ void init_ws_kernel(unsigned int* __restrict__ gHist, int* __restrict__ gCount) {
  int i = blockIdx.x * blockDim.x + threadIdx.x;
  if (i < NBATCH * NCLS * 256) gHist[i] = 0u;
  if (i < NBATCH * NCLS) gCount[i] = 0;
}

// K1 (PHASE 0): slab-streamed per-class 256-bin histograms -> global hist.
// K3 (PHASE 1): slab-streamed collect of key8 >= T8[c] candidates -> lists.
// One block per (image, slab); TDM streams contiguous 48-anchor tiles.
template <int PHASE>
__global__ __launch_bounds__(256)
void slab_kernel(const float* __restrict__ pred,
                 unsigned int* __restrict__ gHist,
                 const int* __restrict__ gT8,
                 int* __restrict__ gCount,
                 float* __restrict__ gListV,
                 int* __restrict__ gListI) {
  __shared__ float stage[2][CHUNKF];
  __shared__ unsigned int lhist[NCLS * 128];   // packed 2x16-bit counters
  int* lT8 = (int*)lhist;                      // PHASE 1 alias (first 80 ints)

  const int tid = threadIdx.x;
  const int b = blockIdx.x / SLABS;
  const int s = blockIdx.x % SLABS;
  const int slabStart = s * SLABR;
  int rows = NANCH - slabStart; if (rows > SLABR) rows = SLABR;
  const float* gbase = pred + ((size_t)b * NANCH + (size_t)slabStart) * NFIELD;

  const uint32_t so0 = (uint32_t)(uintptr_t)(&stage[0][0]);  // LDS byte offset
  const uint32_t so1 = (uint32_t)(uintptr_t)(&stage[1][0]);

  if (PHASE == 0) {
    for (int i = tid; i < NCLS * 128; i += 256) lhist[i] = 0u;
  } else {
    for (int i = tid; i < NCLS; i += 256) lT8[i] = gT8[b * NCLS + i];
  }

  const int nch = (rows + CROWS - 1) / CROWS;
#if USE_TDM
  if (tid < 32) {
    int r0 = rows < CROWS ? rows : CROWS;
    tdm_issue_tile(so0, gbase, r0 * NFIELD);
  }
#endif
  for (int k = 0; k < nch; ++k) {
    const int rbase = k * CROWS;
    int rrows = rows - rbase; if (rrows > CROWS) rrows = CROWS;
    const int buf = k & 1;
#if USE_TDM
    if (tid < 32) {                      // wave 0 drives the DMA pipeline
      const int nb = rbase + CROWS;
      if (nb < rows) {
        int nr = rows - nb; if (nr > CROWS) nr = CROWS;
        tdm_issue_tile(buf ? so0 : so1, gbase + (size_t)nb * NFIELD, nr * NFIELD);
        __builtin_amdgcn_s_wait_tensorcnt(1);  // in-order: chunk k resident
      } else {
        __builtin_amdgcn_s_wait_tensorcnt(0);
      }
    }
#else
    for (int i = tid; i < rrows * NFIELD; i += 256)
      stage[buf][i] = gbase[(size_t)rbase * NFIELD + i];
#endif
    __syncthreads();
    // Threads 0..251: 3 anchor rows per sweep, fixed field (=> fixed class).
    if (tid < 252) {
      const int fRow = tid / 84;
      const int f = tid % 84;
      const int c = f - 4;
      if (c >= 0) {
        for (int g = 0; g < rrows; g += 3) {
          const int r = g + fRow;
          if (r < rrows) {
            float v = stage[buf][r * NFIELD + f];
            if (v > LOGIT_TH) {
              unsigned int k8 = fkey(v) >> 24;
              if (PHASE == 0) {
                atomicAdd(&lhist[(unsigned)c * 128u + (k8 >> 1)],
                          (k8 & 1u) ? 0x10000u : 1u);
              } else {
                if ((int)k8 >= lT8[c]) {
                  int bc = b * NCLS + c;
                  int p = atomicAdd(&gCount[bc], 1);
                  if (p < LISTCAP) {
                    gListV[(size_t)bc * LISTCAP + p] = v;
                    gListI[(size_t)bc * LISTCAP + p] = slabStart + rbase + r;
                  }
                }
              }
            }
          }
        }
      }
    }
    __syncthreads();
  }

  if (PHASE == 0) {
    const size_t hb = (size_t)b * NCLS * 256;
    for (int i = tid; i < NCLS * 128; i += 256) {
      unsigned int hv = lhist[i];
      if (hv) {
        unsigned int lo = hv & 0xFFFFu, hi = hv >> 16;
        if (lo) atomicAdd(&gHist[hb + (size_t)i * 2], lo);
        if (hi) atomicAdd(&gHist[hb + (size_t)i * 2 + 1], hi);
      }
    }
  }
}

// K2: coarse per-(b,c) threshold bin (bin of the 100th largest logit).
__global__ void t8_kernel(const unsigned int* __restrict__ gHist, int* __restrict__ gT8) {
  int bc = blockIdx.x * blockDim.x + threadIdx.x;
  if (bc >= NBATCH * NCLS) return;
  const unsigned int* h = gHist + (size_t)bc * 256;
  int cum = 0, T = 0;                 // T=0 => collect everything (total<100)
  for (int i = 255; i >= 0; --i) {
    cum += (int)h[i];
    if (cum >= KPER) { T = i; break; }
  }
  gT8[bc] = T;
}

// K4: one block per (b,c): exact top-100 from the candidate list, greedy NMS,
// write per-class candidates.
__global__ __launch_bounds__(256)
void class_select_kernel(const float* __restrict__ pred,
                         const int* __restrict__ gCount,
                         const float* __restrict__ gListV,
                         const int* __restrict__ gListI,
                         float* __restrict__ candScores,
                         float* __restrict__ candBoxes) {
  __shared__ float listS[LISTCAP];
  __shared__ int   listIdx[LISTCAP];
  __shared__ unsigned int hist[NBINS];
  __shared__ unsigned int superS[64];
  __shared__ float binS[LISTCAP];
  __shared__ int   binI[LISTCAP];
  __shared__ float selS[128];
  __shared__ int   selI[128];
  __shared__ float fS[FCAP];
  __shared__ int   fI[FCAP];
  __shared__ float wS[8];
  __shared__ int   wN[8];
  __shared__ int   wP[8];
  __shared__ int   ctrs[4];
  __shared__ float bxs[KPER][4];
  __shared__ int   keepS[KPER];

  const int tid = threadIdx.x;
  const int bc = blockIdx.x;
  const int b = bc / NCLS;
  int cnt = gCount[bc]; if (cnt > LISTCAP) cnt = LISTCAP;
  const size_t lbase = (size_t)bc * LISTCAP;
  for (int i = tid; i < cnt; i += 256) { listS[i] = gListV[lbase + i]; listIdx[i] = gListI[lbase + i]; }
  for (int i = tid; i < NBINS; i += 256) hist[i] = 0u;
  if (tid < 4) ctrs[tid] = 0;
  __syncthreads();

  for (int i = tid; i < cnt; i += 256) atomicAdd(&hist[fkey(listS[i]) >> 21], 1u);
  __syncthreads();
  const int T = find_threshold_bin(hist, superS, KPER, tid, &ctrs[2]);
  if (tid == 0) { ctrs[0] = 0; ctrs[1] = 0; }
  __syncthreads();
  for (int i = tid; i < cnt; i += 256) {
    float v = listS[i]; int n = listIdx[i];
    int bin = (int)(fkey(v) >> 21);
    if (bin > T) {
      int p = atomicAdd(&ctrs[0], 1);
      if (p < 128) { selS[p] = v; selI[p] = n; }
    } else if (bin == T) {
      int p = atomicAdd(&ctrs[1], 1);
      binS[p] = v; binI[p] = n;              // p < cnt <= LISTCAP
    }
  }
  __syncthreads();
  const int G = ctrs[0];
  const int M = ctrs[1];
  const int total = select_from_bin(binS, binI, M, G, KPER - G, hist, superS,
                                    selS, selI, fS, fI, wS, wN, wP, ctrs, tid);
  for (int p = total + tid; p < 128; p += 256) { selS[p] = PAD_VAL; selI[p] = 0x7FFFFFFF; }
  __syncthreads();

  sort128_desc(selS, selI, tid);

  // Decode boxes for top-100 (analytic RetinaNet anchors).
  if (tid < KPER) {
    float sv = selS[tid];
    bool real = sv > REAL_MIN;
    float x1 = 0.f, y1 = 0.f, x2 = 0.f, y2 = 0.f;
    if (real) {
      int n = selI[tid];
      int rem = n, lvl = 0;
      if (rem >= 57600) { rem -= 57600; lvl = 1;
        if (rem >= 14400) { rem -= 14400; lvl = 2;
          if (rem >= 3600) { rem -= 3600; lvl = 3;
            if (rem >= 900) { rem -= 900; lvl = 4; } } } }
      const int   fwArr[5]     = {80, 40, 20, 10, 5};
      const float strideArr[5] = {8.f, 16.f, 32.f, 64.f, 128.f};
      const float areaArr[5]   = {1024.f, 4096.f, 16384.f, 65536.f, 262144.f};
      int a = rem % 9;
      int cell = rem / 9;
      int gx = cell % fwArr[lvl];
      int gy = cell / fwArr[lvl];
      float st  = strideArr[lvl];
      float acx = ((float)gx + 0.5f) * st;
      float acy = ((float)gy + 0.5f) * st;
      int ridx = a / 3, sidx = a % 3;
      float ratio = (ridx == 0) ? 0.5f : ((ridx == 1) ? 1.0f : 2.0f);
      float ahb = sqrtf(areaArr[lvl] / ratio);
      float awb = areaArr[lvl] / ahb;
      float sc  = (sidx == 0) ? 1.0f
                : ((sidx == 1) ? 1.2599210498948732f : 1.5874010519681994f);
      float anw = sc * awb, anh = sc * ahb;
      const float* pr = pred + ((size_t)b * NANCH + (size_t)n) * NFIELD;
      float t0 = pr[0] * 0.1f, t1 = pr[1] * 0.1f;
      float t2 = pr[2] * 0.2f, t3 = pr[3] * 0.2f;
      float px = t0 * anw + acx;
      float py = t1 * anh + acy;
      float pw = expf(t2) * anw;
      float ph = expf(t3) * anh;
      x1 = px - 0.5f * pw; y1 = py - 0.5f * ph;
      x2 = px + 0.5f * pw; y2 = py + 0.5f * ph;
    }
    bxs[tid][0] = x1; bxs[tid][1] = y1; bxs[tid][2] = x2; bxs[tid][3] = y2;
    keepS[tid] = real ? 1 : 0;
  }
  __syncthreads();

  // Greedy NMS, IoU > 0.5, score-descending order.
  for (int i = 0; i < KPER; ++i) {
    if (keepS[i] && tid < KPER && tid > i) {
      float ltx = fmaxf(bxs[i][0], bxs[tid][0]);
      float lty = fmaxf(bxs[i][1], bxs[tid][1]);
      float rbx = fminf(bxs[i][2], bxs[tid][2]);
      float rby = fminf(bxs[i][3], bxs[tid][3]);
      float iw = fmaxf(rbx - ltx, 0.f);
      float ih = fmaxf(rby - lty, 0.f);
      float inter = iw * ih;
      float a1 = (bxs[i][2] - bxs[i][0]) * (bxs[i][3] - bxs[i][1]);
      float a2 = (bxs[tid][2] - bxs[tid][0]) * (bxs[tid][3] - bxs[tid][1]);
      float iou = inter / (a1 + a2 - inter + 1e-8f);
      if (iou > 0.5f) keepS[tid] = 0;
    }
    __syncthreads();
  }

  if (tid < KPER) {
    size_t o = (size_t)bc * KPER + (size_t)tid;
    bool real = selS[tid] > REAL_MIN;
    float outs = (real && keepS[tid]) ? (1.0f / (1.0f + expf(-selS[tid]))) : -1.0f;
    candScores[o] = outs;
    candBoxes[o * 4 + 0] = bxs[tid][0];
    candBoxes[o * 4 + 1] = bxs[tid][1];
    candBoxes[o * 4 + 2] = bxs[tid][2];
    candBoxes[o * 4 + 3] = bxs[tid][3];
  }
}

// K5: one block per image: exact top-100 over 8000 candidates, outputs.
__global__ __launch_bounds__(256)
void final_topk_kernel(const float* __restrict__ candScores,
                       const float* __restrict__ candBoxes,
                       float* __restrict__ out) {
  const int b = blockIdx.x;
  const int tid = threadIdx.x;
  __shared__ float s2[NCLS * KPER];
  __shared__ unsigned int hist[NBINS];
  __shared__ unsigned int superS[64];
  __shared__ float binS[BINCAP2];
  __shared__ int   binI[BINCAP2];
  __shared__ float selS[128];
  __shared__ int   selI[128];
  __shared__ float fS[FCAP];
  __shared__ int   fI[FCAP];
  __shared__ float wS[8];
  __shared__ int   wN[8];
  __shared__ int   wP[8];
  __shared__ int   ctrs[4];

  for (int i = tid; i < NCLS * KPER; i += 256)
    s2[i] = candScores[(size_t)b * NCLS * KPER + i];
  for (int i = tid; i < NBINS; i += 256) hist[i] = 0u;
  if (tid < 4) ctrs[tid] = 0;
  __syncthreads();

  for (int i = tid; i < NCLS * KPER; i += 256)
    atomicAdd(&hist[fkey(s2[i]) >> 21], 1u);
  __syncthreads();

  const int T = find_threshold_bin(hist, superS, MAXDET, tid, &ctrs[2]);
  if (tid == 0) { ctrs[0] = 0; ctrs[1] = 0; }
  __syncthreads();

  for (int i = tid; i < NCLS * KPER; i += 256) {
    float v = s2[i];
    int bin = (int)(fkey(v) >> 21);
    if (bin > T) {
      int p = atomicAdd(&ctrs[0], 1);
      if (p < 128) { selS[p] = v; selI[p] = i; }
    } else if (bin == T) {
      int p = atomicAdd(&ctrs[1], 1);
      if (p < BINCAP2) { binS[p] = v; binI[p] = i; }
    }
  }
  __syncthreads();

  const int G = ctrs[0];
  int M = ctrs[1]; if (M > BINCAP2) M = BINCAP2;
  const int total = select_from_bin(binS, binI, M, G, MAXDET - G, hist, superS,
                                    selS, selI, fS, fI, wS, wN, wP, ctrs, tid);
  for (int p = total + tid; p < 128; p += 256) { selS[p] = PAD_VAL; selI[p] = 0x7FFFFFFF; }
  __syncthreads();

  sort128_desc(selS, selI, tid);

  if (tid < MAXDET) {
    float ts = selS[tid];
    int q = selI[tid];
    bool good = ts > 0.0f;
    float* fb4 = out + ((size_t)b * MAXDET + (size_t)tid) * 4;
    if (good) {
      size_t cbo = ((size_t)b * NCLS * KPER + (size_t)q) * 4;
      fb4[0] = candBoxes[cbo + 0];
      fb4[1] = candBoxes[cbo + 1];
      fb4[2] = candBoxes[cbo + 2];
      fb4[3] = candBoxes[cbo + 3];
    } else {
      fb4[0] = 0.f; fb4[1] = 0.f; fb4[2] = 0.f; fb4[3] = 0.f;
    }
    out[(size_t)NBATCH * MAXDET * 4 + (size_t)b * MAXDET + tid] = good ? ts : 0.f;
    out[(size_t)NBATCH * MAXDET * 5 + (size_t)b * MAXDET + tid] =
        good ? (float)(q / KPER) : 0.f;
  }
  int vc = __syncthreads_count(tid < MAXDET && selS[tid] > 0.0f);
  if (tid == 0)
    ((int*)out)[NBATCH * MAXDET * 6 + b] = vc;
}

extern "C" void kernel_launch(void* const* d_in, const int* in_sizes, int n_in,
                              void* d_out, int out_size, void* d_ws, size_t ws_size,
                              hipStream_t stream) {
  (void)in_sizes; (void)n_in; (void)out_size; (void)ws_size;
  // d_in[0] = images (unused by the reference math), d_in[1] = predictions.
  const float* pred = (const float*)d_in[1];

  // Workspace layout (all 4-byte elements, ~12.4 MB total).
  uint8_t* w = (uint8_t*)d_ws;
  unsigned int* gHist = (unsigned int*)w; w += (size_t)NBATCH * NCLS * 256 * 4;
  int* gT8 = (int*)w;                     w += (size_t)NBATCH * NCLS * 4;
  int* gCount = (int*)w;                  w += (size_t)NBATCH * NCLS * 4;
  float* gListV = (float*)w;              w += (size_t)NBATCH * NCLS * LISTCAP * 4;
  int* gListI = (int*)w;                  w += (size_t)NBATCH * NCLS * LISTCAP * 4;
  float* candScores = (float*)w;          w += (size_t)NBATCH * NCLS * KPER * 4;
  float* candBoxes = (float*)w;

  const int initN = NBATCH * NCLS * 256;
  init_ws_kernel<<<(initN + 255) / 256, 256, 0, stream>>>(gHist, gCount);
  slab_kernel<0><<<NBATCH * SLABS, 256, 0, stream>>>(pred, gHist, gT8, gCount, gListV, gListI);
  t8_kernel<<<(NBATCH * NCLS + 127) / 128, 128, 0, stream>>>(gHist, gT8);
  slab_kernel<1><<<NBATCH * SLABS, 256, 0, stream>>>(pred, gHist, gT8, gCount, gListV, gListI);
  class_select_kernel<<<NBATCH * NCLS, 256, 0, stream>>>(pred, gCount, gListV, gListI,
                                                         candScores, candBoxes);
  final_topk_kernel<<<NBATCH, 256, 0, stream>>>(candScores, candBoxes, (float*)d_out);
}